// FFTConv_42167988912397
// MI455X (gfx1250) — compile-verified
//
#include <hip/hip_runtime.h>

// ---------------------------------------------------------------------------
// FFT circular convolution (Hyena/SSM style) for MI455X (gfx1250, wave32).
//   y = tanh( ifft(fft(u) * fft(k)).real + diag(D) * u )
// One workgroup (512 threads = 16 waves) per (b,h) row of length 16384.
// Whole row resident in LDS (2 ping-pong interleaved-complex buffers =
// 256 KB of the 320 KB WGP LDS). Stockham autosort FFT, radices [16,16,16,4].
// Radix-16 sub-DFTs run on the matrix pipes via V_WMMA_F32_16X16X4_F32
// (fp32 WMMA -> full FFT accuracy). Twiddles use the native TRANS pipe
// (v_sin_f32 / v_cos_f32) which co-issues with VALU/WMMA.
// ---------------------------------------------------------------------------

typedef __attribute__((ext_vector_type(2))) float v2f;
typedef __attribute__((ext_vector_type(8))) float v8f;

#define LFFT     16384
#define NTHREADS 512
#define TWO_PI   6.28318530717958647692f

__device__ __forceinline__ v8f wmma_f32(v2f a, v2f b, v8f c) {
  // D(16x16,f32) = A(16x4,f32) x B(4x16,f32) + C
  return __builtin_amdgcn_wmma_f32_16x16x4_f32(
      /*neg_a=*/false, a, /*neg_b=*/false, b,
      /*c_mod=*/(short)0, c, /*reuse_a=*/false, /*reuse_b=*/false);
}

// tanh via native v_exp_f32 / v_rcp_f32 (saturates to +-1 correctly).
__device__ __forceinline__ float fast_tanh(float x) {
  float t = __builtin_amdgcn_exp2f(x * 2.88539008177792681472f); // exp(2x)
  float r = __builtin_amdgcn_rcpf(t + 1.0f);
  return 1.0f - 2.0f * r;
}

// One Stockham radix-16 stage on interleaved-complex LDS data:
//   dst[expand(j)] = DFT16( twiddle * src[j + r*N/16] )
// F-matrix (A operand) layout per ISA: lane L holds M = L&15; VGPR pair holds
// K = {4kk + 2*(L>>4), +1}. Data (B operand) mirrored: lane L holds N = L&15,
// same K mapping. D layout: VGPR v holds M = v + 8*(L>>4), N = L&15.
__device__ __forceinline__ void radix16_stage(
    const float2* __restrict__ src, float2* __restrict__ dst,
    int Ns, int nsLog, float sgn,
    const v2f* fre, const v2f* fimA, const v2f* fimB,  // fimA = Fim, fimB = -Fim
    int lane, int wave)
{
  const int NR   = LFFT / 16;          // 1024 sub-transforms
  const int col  = lane & 15;
  const int half = lane >> 4;
  const float wscale = sgn * TWO_PI / (float)(Ns * 16);

#pragma unroll
  for (int tt = 0; tt < 4; ++tt) {     // 64 tiles / 16 waves = 4 tiles per wave
    const int j  = (wave + 16 * tt) * 16 + col;   // sub-transform index (= B column)
    const int jm = j & (Ns - 1);
    const float th = wscale * (float)jm;          // per-(tile,lane) base angle
    v8f accRe = {0.f,0.f,0.f,0.f,0.f,0.f,0.f,0.f};
    v8f accIm = {0.f,0.f,0.f,0.f,0.f,0.f,0.f,0.f};

#pragma unroll
    for (int kk = 0; kk < 4; ++kk) {   // K-chunks of the 16x16 product
      const int r0 = 4 * kk + 2 * half;
      float2 z0 = src[j + r0 * NR];        // ds_load_b64
      float2 z1 = src[j + (r0 + 1) * NR];
      if (Ns > 1) {                    // uniform branch (folds after inlining)
        float a0 = th * (float)r0;     // |a| < 2*pi -> native range OK
        float a1 = a0 + th;
        float c0 = __builtin_amdgcn_cosf(a0), s0 = __builtin_amdgcn_sinf(a0);
        float c1 = __builtin_amdgcn_cosf(a1), s1 = __builtin_amdgcn_sinf(a1);
        float t;
        t = z0.x * c0 - z0.y * s0;  z0.y = z0.x * s0 + z0.y * c0;  z0.x = t;
        t = z1.x * c1 - z1.y * s1;  z1.y = z1.x * s1 + z1.y * c1;  z1.x = t;
      }
      v2f bre = {z0.x, z1.x};
      v2f bim = {z0.y, z1.y};
      // complex product: Re = Fre*Vre - Fim*Vim ; Im = Fre*Vim + Fim*Vre
      accRe = wmma_f32(fre[kk],  bre, accRe);
      accRe = wmma_f32(fimB[kk], bim, accRe);
      accIm = wmma_f32(fre[kk],  bim, accIm);
      accIm = wmma_f32(fimA[kk], bre, accIm);
    }

    const int idxD = ((j >> nsLog) << (nsLog + 4)) + jm;  // Stockham expand(j,Ns,16)
#pragma unroll
    for (int v = 0; v < 8; ++v) {
      const int rp = v + 8 * half;     // output row r'
      dst[idxD + rp * Ns] = make_float2(accRe[v], accIm[v]);  // ds_store_b64
    }
  }
}

// Final Stockham radix-4 stage (Ns = 4096), scalar VALU + TRANS twiddles.
__device__ __forceinline__ void radix4_stage(
    const float2* __restrict__ src, float2* __restrict__ dst,
    int tid, float sgn)
{
  const int Q = LFFT / 4;              // 4096
  const float wscale = sgn * TWO_PI / (float)LFFT;
#pragma unroll
  for (int m = 0; m < 8; ++m) {
    const int j = m * NTHREADS + tid;  // 0..4095 ; expand(j,4096,4) = j
    const float th = wscale * (float)j;
    float ar[4], ai[4];
#pragma unroll
    for (int r = 0; r < 4; ++r) {
      float2 z = src[j + r * Q];
      ar[r] = z.x; ai[r] = z.y;
    }
#pragma unroll
    for (int r = 1; r < 4; ++r) {
      float a = th * (float)r;         // |a| < 4.72 rad
      float c = __builtin_amdgcn_cosf(a), s = __builtin_amdgcn_sinf(a);
      float t = ar[r] * c - ai[r] * s;
      ai[r] = ar[r] * s + ai[r] * c;
      ar[r] = t;
    }
    float t0r = ar[0] + ar[2], t0i = ai[0] + ai[2];
    float t1r = ar[0] - ar[2], t1i = ai[0] - ai[2];
    float t2r = ar[1] + ar[3], t2i = ai[1] + ai[3];
    float t3r = ar[1] - ar[3], t3i = ai[1] - ai[3];
    dst[j + 0 * Q] = make_float2(t0r + t2r,       t0i + t2i);
    dst[j + 2 * Q] = make_float2(t0r - t2r,       t0i - t2i);
    dst[j + 1 * Q] = make_float2(t1r - sgn * t3i, t1i + sgn * t3r);
    dst[j + 3 * Q] = make_float2(t1r + sgn * t3i, t1i - sgn * t3r);
  }
}

__global__ void __launch_bounds__(NTHREADS)
fftconv_kernel(const float* __restrict__ U, const float* __restrict__ Kf,
               const float* __restrict__ Dm, float* __restrict__ Y, int Hn)
{
  __shared__ __align__(16) float2 sA[LFFT];   // 128 KB
  __shared__ __align__(16) float2 sB[LFFT];   // 128 KB

  const int blk  = blockIdx.x;         // row = b*H + h
  const int h    = blk % Hn;
  const float* __restrict__ urow = U  + (size_t)blk * LFFT;
  const float* __restrict__ krow = Kf + (size_t)h   * LFFT;

  const int tid  = threadIdx.x;
  const int lane = tid & 31;
  const int wave = tid >> 5;

  // Coalesced load; pack z = u + i*k interleaved in LDS (b128 stores);
  // keep u in registers for the feedthrough epilogue.
  float4 ureg[8];
#pragma unroll
  for (int m = 0; m < 8; ++m) {
    int i4 = m * NTHREADS + tid;
    float4 uv = ((const float4*)urow)[i4];
    float4 kv = ((const float4*)krow)[i4];
    ureg[m] = uv;
    ((float4*)sA)[2 * i4 + 0] = make_float4(uv.x, kv.x, uv.y, kv.y);
    ((float4*)sA)[2 * i4 + 1] = make_float4(uv.z, kv.z, uv.w, kv.w);
  }

  // DFT-16 matrix in WMMA A-layout (forward sign: F = exp(-2*pi*i*m*r/16)).
  v2f fre[4], fim[4], fimn[4];
  {
    const int m16 = lane & 15, half = lane >> 4;
#pragma unroll
    for (int kk = 0; kk < 4; ++kk) {
      int r0 = 4 * kk + 2 * half;
      float a0 = -TWO_PI * (float)((m16 * r0)       & 15) * (1.0f / 16.0f);
      float a1 = -TWO_PI * (float)((m16 * (r0 + 1)) & 15) * (1.0f / 16.0f);
      float c0 = __builtin_amdgcn_cosf(a0), s0 = __builtin_amdgcn_sinf(a0);
      float c1 = __builtin_amdgcn_cosf(a1), s1 = __builtin_amdgcn_sinf(a1);
      fre [kk] = { c0,  c1};
      fim [kk] = { s0,  s1};
      fimn[kk] = {-s0, -s1};
    }
  }
  __syncthreads();

  // ---- forward FFT of z = u + i*k (sgn = -1) ----
  radix16_stage(sA, sB,   1, 0, -1.f, fre, fim,  fimn, lane, wave);
  __syncthreads();
  radix16_stage(sB, sA,  16, 4, -1.f, fre, fim,  fimn, lane, wave);
  __syncthreads();
  radix16_stage(sA, sB, 256, 8, -1.f, fre, fim,  fimn, lane, wave);
  __syncthreads();
  radix4_stage (sB, sA, tid, -1.f);
  __syncthreads();

  // ---- Hermitian split of packed spectrum + pointwise product ----
  // U = (Z[f]+conj(Z[-f]))/2 ; K = (Z[f]-conj(Z[-f]))/(2i) ; Y = U*K, Y[-f]=conj(Y[f])
  for (int f = tid; f <= LFFT / 2; f += NTHREADS) {
    int g = (LFFT - f) & (LFFT - 1);
    float2 zf = sA[f];
    float2 zg = sA[g];
    float ur = 0.5f * (zf.x + zg.x), ui = 0.5f * (zf.y - zg.y);
    float kr = 0.5f * (zf.y + zg.y), ki = 0.5f * (zg.x - zf.x);
    float yr = ur * kr - ui * ki;
    float yi = ur * ki + ui * kr;
    sA[f] = make_float2(yr,  yi);
    sA[g] = make_float2(yr, -yi);
  }
  __syncthreads();

  // ---- inverse FFT (sgn = +1; Fim sign flips => swap fim/fimn) ----
  radix16_stage(sA, sB,   1, 0, +1.f, fre, fimn, fim, lane, wave);
  __syncthreads();
  radix16_stage(sB, sA,  16, 4, +1.f, fre, fimn, fim, lane, wave);
  __syncthreads();
  radix16_stage(sA, sB, 256, 8, +1.f, fre, fimn, fim, lane, wave);
  __syncthreads();
  radix4_stage (sB, sA, tid, +1.f);
  __syncthreads();

  // ---- epilogue: y/N + diag(D)*u, fast tanh, coalesced store ----
  const float dd   = Dm[(size_t)h * Hn + h];
  const float invN = 1.0f / (float)LFFT;
  float* __restrict__ orow = Y + (size_t)blk * LFFT;
#pragma unroll
  for (int m = 0; m < 8; ++m) {
    int i4 = m * NTHREADS + tid;
    float4 q0 = ((const float4*)sA)[2 * i4 + 0];  // {re0, im0, re1, im1}
    float4 q1 = ((const float4*)sA)[2 * i4 + 1];  // {re2, im2, re3, im3}
    float4 uv = ureg[m];
    float4 o;
    o.x = fast_tanh(q0.x * invN + dd * uv.x);
    o.y = fast_tanh(q0.z * invN + dd * uv.y);
    o.z = fast_tanh(q1.x * invN + dd * uv.z);
    o.w = fast_tanh(q1.z * invN + dd * uv.w);
    ((float4*)orow)[i4] = o;
  }
}

extern "C" void kernel_launch(void* const* d_in, const int* in_sizes, int n_in,
                              void* d_out, int out_size, void* d_ws, size_t ws_size,
                              hipStream_t stream) {
  (void)n_in; (void)d_ws; (void)ws_size;
  const float* u  = (const float*)d_in[0];   // (B,H,L) f32
  const float* k  = (const float*)d_in[1];   // (H,L)   f32
  const float* Dm = (const float*)d_in[2];   // (H,H)   f32
  float* out = (float*)d_out;                // (B,H,L) f32

  const int rows = out_size / LFFT;          // B*H = 2048
  const int Hn   = in_sizes[1] / LFFT;       // 256

  fftconv_kernel<<<dim3(rows), dim3(NTHREADS), 0, stream>>>(u, k, Dm, out, Hn);
}